// AttnDecoderRNN_52295521796629
// MI455X (gfx1250) — compile-verified
//
#include <hip/hip_runtime.h>
#include <math.h>

// ---------------------------------------------------------------------------
// AttnDecoderRNN single step, batch 1. Bandwidth-bound GEMV chain (~238 MB of
// fp32 weights -> ~10.2 us floor at 23.3 TB/s). All large GEMVs are carried by
// v_wmma_f32_16x16x4_f32: one wave computes 16 output rows, A = 16x4 tile of W
// (per CDNA5 A-layout), B = x chunk broadcast across all 16 columns.
// ---------------------------------------------------------------------------

typedef float v2f __attribute__((ext_vector_type(2)));
typedef float v8f __attribute__((ext_vector_type(8)));

#define NHID 1024
#define NOUT 50257
#define MAXLEN 24

__device__ __forceinline__ float waveReduceSum(float v) {
#pragma unroll
  for (int off = 16; off > 0; off >>= 1) v += __shfl_xor(v, off, 32);
  return v;
}

// ---------------------------------------------------------------------------
// Stage 0: embedding lookup, attention scores + softmax, attention-applied
// context; writes cat1 = [embedded, attn_applied] (2048 floats) and the
// attn_weights output (24 floats). One block of 1024 threads (32 waves).
// ---------------------------------------------------------------------------
__global__ void k_prep(const int* __restrict__ inp, const float* __restrict__ hidden,
                       const float* __restrict__ enc_outs, const float* __restrict__ emb_W,
                       const float* __restrict__ attn_W, const float* __restrict__ attn_b,
                       float* __restrict__ cat1, float* __restrict__ attn_out) {
  __shared__ float sE[NHID];
  __shared__ float sH[NHID];
  __shared__ float sS[MAXLEN];
  __shared__ float sAW[MAXLEN];
  const int t = threadIdx.x;  // 1024 threads
  const int tok = inp[0];
  float e = emb_W[(size_t)tok * NHID + t];
  sE[t] = e;
  cat1[t] = e;
  sH[t] = hidden[t];
  __syncthreads();

  // 24 waves compute the 24 attention scores over the 2048-wide concat input.
  const int w = t >> 5, lane = t & 31;
  if (w < MAXLEN) {
    float p = 0.f;
    const float* __restrict__ wr = attn_W + (size_t)w * 2 * NHID;
    for (int j = lane; j < 2 * NHID; j += 32) {
      float xin = (j < NHID) ? sE[j] : sH[j - NHID];
      p += xin * wr[j];
    }
    p = waveReduceSum(p);
    if (lane == 0) sS[w] = p + attn_b[w];
  }
  __syncthreads();

  if (t == 0) {  // softmax over 24 values: trivial serial work
    float m = sS[0];
    for (int i = 1; i < MAXLEN; i++) m = fmaxf(m, sS[i]);
    float s = 0.f;
    for (int i = 0; i < MAXLEN; i++) { float v = expf(sS[i] - m); sAW[i] = v; s += v; }
    float inv = 1.f / s;
    for (int i = 0; i < MAXLEN; i++) { sAW[i] *= inv; attn_out[i] = sAW[i]; }
  }
  __syncthreads();

  // attn_applied[t] = sum_i aw[i] * encoder_outputs[i][t] (coalesced over t)
  float acc = 0.f;
#pragma unroll
  for (int i = 0; i < MAXLEN; i++) acc += sAW[i] * enc_outs[i * NHID + t];
  cat1[NHID + t] = acc;
}

// ---------------------------------------------------------------------------
// WMMA GEMV: y = act(W @ x + bias). W is row-major N x K (K % 4 == 0).
// One wave -> 16 rows, K-loop in steps of 4 through v_wmma_f32_16x16x4_f32.
// x is staged in LDS (dynamic shared, K floats). ACT: 0 = none, 1 = relu.
// ---------------------------------------------------------------------------
template <int ACT>
__global__ void k_wmma_gemv(const float* __restrict__ W, const float* __restrict__ x,
                            const float* __restrict__ bias, float* __restrict__ y,
                            int N, int K) {
  extern __shared__ float sx[];
  for (int i = threadIdx.x; i < K; i += blockDim.x) sx[i] = x[i];
  __syncthreads();

  const int wavesPerBlk = blockDim.x >> 5;
  const int wav = blockIdx.x * wavesPerBlk + (threadIdx.x >> 5);
  const int tiles = (N + 15) >> 4;
  if (wav >= tiles) return;  // whole-wave exit; EXEC stays all-ones for WMMA

  const int lane = threadIdx.x & 31;
  const int m = lane & 15;
  const int s2 = (lane >> 4) << 1;  // lanes 0-15 -> k+0/1 ; lanes 16-31 -> k+2/3
  const int r0 = wav << 4;
  int row = r0 + m;
  if (row > N - 1) row = N - 1;  // clamp: duplicate reads, stores are guarded

  const float* __restrict__ Arow = W + (size_t)row * K + s2;
  v8f c = {0.f, 0.f, 0.f, 0.f, 0.f, 0.f, 0.f, 0.f};

#pragma unroll 4
  for (int k0 = 0; k0 < K; k0 += 4) {
    v2f a = *(const v2f*)(Arow + k0);      // A 16x4 tile (CDNA5 A-layout)
    v2f b = *(const v2f*)(sx + k0 + s2);   // B 4x16: x broadcast to all columns
    c = __builtin_amdgcn_wmma_f32_16x16x4_f32(false, a, false, b, (short)0, c,
                                              false, false);
  }

  // D layout: lane n<16 holds y[r0+0..7] in c[0..7]; lane n>=16 holds y[r0+8..15].
  const int base = r0 + ((lane >> 4) << 3);
  if (lane == 0 || lane == 16) {
    if (base + 8 <= N) {
#pragma unroll
      for (int i = 0; i < 8; i++) {
        float v = c[i] + bias[base + i];
        if (ACT == 1) v = fmaxf(v, 0.f);
        y[base + i] = v;
      }
    } else {
#pragma unroll
      for (int i = 0; i < 8; i++) {
        if (base + i < N) {
          float v = c[i] + bias[base + i];
          if (ACT == 1) v = fmaxf(v, 0.f);
          y[base + i] = v;
        }
      }
    }
  }
}

// ---------------------------------------------------------------------------
// GRU pointwise combine: h = (1-z)*n + z*h_old. One block, 1024 threads.
// ---------------------------------------------------------------------------
__global__ void k_gru(const float* __restrict__ gi, const float* __restrict__ gh,
                      const float* __restrict__ h_old, float* __restrict__ h_new,
                      float* __restrict__ h_out) {
  const int t = threadIdx.x;  // 1024
  float ir = gi[t], iz = gi[NHID + t], inn = gi[2 * NHID + t];
  float hr = gh[t], hz = gh[NHID + t], hnn = gh[2 * NHID + t];
  float r = 1.f / (1.f + expf(-(ir + hr)));
  float z = 1.f / (1.f + expf(-(iz + hz)));
  float n = tanhf(inn + r * hnn);
  float h = (1.f - z) * n + z * h_old[t];
  h_new[t] = h;
  h_out[t] = h;
}

// ---------------------------------------------------------------------------
// Log-sum-exp reduction over the 50257 logits. One block, 1024 threads.
// red[0] = max, red[1] = log(sum exp(x - max)).
// ---------------------------------------------------------------------------
__global__ void k_lse(const float* __restrict__ logits, float* __restrict__ red) {
  __shared__ float sm[32];
  __shared__ float ss[32];
  __shared__ float smax;
  const int t = threadIdx.x;
  float m = -3.402823e38f;
  for (int i = t; i < NOUT; i += 1024) m = fmaxf(m, logits[i]);
#pragma unroll
  for (int off = 16; off > 0; off >>= 1) m = fmaxf(m, __shfl_xor(m, off, 32));
  if ((t & 31) == 0) sm[t >> 5] = m;
  __syncthreads();
  if (t < 32) {
    float v = sm[t];
#pragma unroll
    for (int off = 16; off > 0; off >>= 1) v = fmaxf(v, __shfl_xor(v, off, 32));
    if (t == 0) smax = v;
  }
  __syncthreads();
  const float mx = smax;
  float s = 0.f;
  for (int i = t; i < NOUT; i += 1024) s += expf(logits[i] - mx);
  s = waveReduceSum(s);
  if ((t & 31) == 0) ss[t >> 5] = s;
  __syncthreads();
  if (t == 0) {
    float tot = 0.f;
#pragma unroll
    for (int i = 0; i < 32; i++) tot += ss[i];
    red[0] = mx;
    red[1] = logf(tot);
  }
}

__global__ void k_logp(const float* __restrict__ logits, const float* __restrict__ red,
                       float* __restrict__ out) {
  const int i = blockIdx.x * blockDim.x + threadIdx.x;
  if (i < NOUT) out[i] = logits[i] - red[0] - red[1];
}

// ---------------------------------------------------------------------------
extern "C" void kernel_launch(void* const* d_in, const int* in_sizes, int n_in,
                              void* d_out, int out_size, void* d_ws, size_t ws_size,
                              hipStream_t stream) {
  const int*   inp      = (const int*)d_in[0];
  const float* hidden   = (const float*)d_in[1];
  /* d_in[2] encoder_output: unused by the reference */
  const float* enc_outs = (const float*)d_in[3];
  const float* emb_W    = (const float*)d_in[4];
  const float* attn_W   = (const float*)d_in[5];
  const float* attn_b   = (const float*)d_in[6];
  const float* comb_W   = (const float*)d_in[7];
  const float* comb_b   = (const float*)d_in[8];
  const float* W_ih     = (const float*)d_in[9];
  const float* W_hh     = (const float*)d_in[10];
  const float* b_ih     = (const float*)d_in[11];
  const float* b_hh     = (const float*)d_in[12];
  const float* out_W    = (const float*)d_in[13];
  const float* out_b    = (const float*)d_in[14];

  float* out    = (float*)d_out;
  float* logp   = out;                  // [50257]
  float* h_out  = out + NOUT;           // [1024]
  float* aw_out = out + NOUT + NHID;    // [24]

  float* ws     = (float*)d_ws;
  float* cat1   = ws;                   // 2048: [embedded | attn_applied]
  float* x1     = cat1 + 2 * NHID;      // 1024: relu(comb)
  float* gi     = x1 + NHID;            // 3072
  float* gh     = gi + 3 * NHID;        // 3072
  float* hN     = gh + 3 * NHID;        // 1024
  float* logits = hN + NHID;            // 50257
  float* red    = logits + NOUT;        // 2

  // 0: embedding + attention + context
  k_prep<<<1, 1024, 0, stream>>>(inp, hidden, enc_outs, emb_W, attn_W, attn_b,
                                 cat1, aw_out);

  // 1: x1 = relu(comb_W @ cat1 + comb_b)        (1024 x 2048, 8 MB)
  {
    int tiles = NHID / 16, blocks = (tiles + 7) / 8;
    k_wmma_gemv<1><<<blocks, 256, 2 * NHID * sizeof(float), stream>>>(
        comb_W, cat1, comb_b, x1, NHID, 2 * NHID);
  }
  // 2: gi = W_ih @ x1 + b_ih ; gh = W_hh @ h + b_hh   (each 3072 x 1024, 12 MB)
  {
    int tiles = 3 * NHID / 16, blocks = (tiles + 7) / 8;
    k_wmma_gemv<0><<<blocks, 256, NHID * sizeof(float), stream>>>(
        W_ih, x1, b_ih, gi, 3 * NHID, NHID);
    k_wmma_gemv<0><<<blocks, 256, NHID * sizeof(float), stream>>>(
        W_hh, hidden, b_hh, gh, 3 * NHID, NHID);
  }
  // 3: GRU pointwise -> hN (and h output)
  k_gru<<<1, 1024, 0, stream>>>(gi, gh, hidden, hN, h_out);

  // 4: logits = out_W @ hN + out_b             (50257 x 1024, 206 MB -> dominant)
  {
    int tiles = (NOUT + 15) / 16, blocks = (tiles + 7) / 8;
    k_wmma_gemv<0><<<blocks, 256, NHID * sizeof(float), stream>>>(
        out_W, hN, out_b, logits, NOUT, NHID);
  }
  // 5: log-softmax
  k_lse<<<1, 1024, 0, stream>>>(logits, red);
  k_logp<<<(NOUT + 255) / 256, 256, 0, stream>>>(logits, red, logp);
}